// SingleRoIExtractor3D_8048768713464
// MI455X (gfx1250) — compile-verified
//
#include <hip/hip_runtime.h>
#include <hip/hip_bf16.h>

typedef float v2f __attribute__((ext_vector_type(2)));
typedef float v8f __attribute__((ext_vector_type(8)));

static constexpr int R_  = 32;    // rois
static constexpr int C_  = 2048;  // channels
static constexpr int NB  = 4;     // batch
static constexpr int T_  = 8;     // temporal
static constexpr int HW  = 256;   // 16*16 spatial
static constexpr int CB  = 16;    // channel blocks per roi
static constexpr int CHB = C_ / CB;   // 128 channels / block
static constexpr int CHW = CHB / 8;   // 16 channels / wave (8 waves/block)

// ---------------------------------------------------------------------------
// Kernel 1: temporal mean  feat(4,2048,8,16,16) -> feat_p(4,2048,256)
// Pure bandwidth: float4 loads, 8 strided reads, 1 write per thread.
// ---------------------------------------------------------------------------
__global__ __launch_bounds__(256) void tmean_kernel(const float* __restrict__ feat,
                                                    float* __restrict__ featp) {
    const int v  = blockIdx.x * 256 + threadIdx.x;  // 0 .. 524287
    const int s4 = v & 63;                          // float4 index within 256
    const int bc = v >> 6;                          // (b*2048 + c)
    const float4* src = (const float4*)feat + (size_t)bc * (T_ * 64) + s4;
    float4 a = src[0];
#pragma unroll
    for (int t = 1; t < T_; ++t) {
        float4 x = src[(size_t)t * 64];
        a.x += x.x; a.y += x.y; a.z += x.z; a.w += x.w;
    }
    const float inv = 1.0f / (float)T_;
    a.x *= inv; a.y *= inv; a.z *= inv; a.w *= inv;
    ((float4*)featp)[(size_t)bc * 64 + s4] = a;
}

// ---------------------------------------------------------------------------
// Kernel 2: RoIAlign as separable WMMA:  Out_c = Wy * F_c * Wx^T   (all 16x16 f32)
// One workgroup = (roi, channel-block). Wy/Wx built once in LDS.
// ---------------------------------------------------------------------------
__global__ __launch_bounds__(256) void roi_align_wmma_kernel(
        const float* __restrict__ featp,
        const float* __restrict__ rois,
        float* __restrict__ out) {
    __shared__ float sWy[256];   // [py][sy] row-major
    __shared__ float sWx[256];   // [px][sx] row-major

    const int r   = blockIdx.x / CB;
    const int cbk = blockIdx.x % CB;
    const int tid = threadIdx.x;

    // ---- build separable interpolation weights (exactly the reference math) ----
    const float scale = 1.0f / 16.0f;
    const float rx1 = rois[r * 5 + 1] * scale - 0.5f;
    const float ry1 = rois[r * 5 + 2] * scale - 0.5f;
    const float rx2 = rois[r * 5 + 3] * scale - 0.5f;
    const float ry2 = rois[r * 5 + 4] * scale - 0.5f;
    const float bh = (ry2 - ry1) * (1.0f / 16.0f);
    const float bw = (rx2 - rx1) * (1.0f / 16.0f);

    const int p = tid >> 4;   // output row/col index
    const int s = tid & 15;   // source feature row/col index

    float wy = 0.0f, wx = 0.0f;
#pragma unroll
    for (int i = 0; i < 2; ++i) {
        const float g = ((float)(2 * p + i) + 0.5f) * 0.5f;  // (i+0.5)/ratio
        {   // y axis sample; 0.5 = sqrt of the 1/4 sample average, split per axis
            const float yy = ry1 + g * bh;
            const float v  = (yy >= -1.0f && yy <= 16.0f) ? 0.5f : 0.0f;
            const float y  = fminf(fmaxf(yy, 0.0f), 15.0f);
            const float y0f = floorf(y);
            const int   y0  = (int)y0f;
            const int   y1i = min(y0 + 1, 15);
            const float ly  = y - y0f;
            const float w   = ((s == y0) ? (1.0f - ly) : 0.0f) + ((s == y1i) ? ly : 0.0f);
            wy += v * w;
        }
        {   // x axis sample
            const float xx = rx1 + g * bw;
            const float v  = (xx >= -1.0f && xx <= 16.0f) ? 0.5f : 0.0f;
            const float x  = fminf(fmaxf(xx, 0.0f), 15.0f);
            const float x0f = floorf(x);
            const int   x0  = (int)x0f;
            const int   x1i = min(x0 + 1, 15);
            const float lx  = x - x0f;
            const float w   = ((s == x0) ? (1.0f - lx) : 0.0f) + ((s == x1i) ? lx : 0.0f);
            wx += v * w;
        }
    }
    sWy[tid] = wy;
    sWx[tid] = wx;
    __syncthreads();

    // ---- load constant WMMA operands (reused across all channels) ----
    const int  lane  = tid & 31;
    const int  lrow  = lane & 15;
    const int  lhalf = lane >> 4;
    const bool lo    = (lhalf == 0);

    v2f aWy[4], bWx[4];
#pragma unroll
    for (int k = 0; k < 4; ++k) {
        const int off = lrow * 16 + 4 * k + 2 * lhalf;   // f32 A/B slice layout
        bWx[k] = *(const v2f*)&sWx[off];   // B of GEMM1: Wx^T slices
        aWy[k] = *(const v2f*)&sWy[off];   // A of GEMM2: Wy slices
    }

    const int b = (int)rois[r * 5 + 0];
    const float* fb = featp + ((size_t)b * C_) * HW;
    float*       ob = out   + ((size_t)r * C_) * HW;
    const int wave = tid >> 5;
    const int c0   = cbk * CHB + wave * CHW;

    for (int cc = 0; cc < CHW; ++cc) {
        const int c = c0 + cc;
        const float* f = fb + (size_t)c * HW;
        if (cc + 1 < CHW) __builtin_prefetch(fb + (size_t)(c + 1) * HW, 0, 1);

        // A of GEMM1: F_c 16x4 slices, per-lane b64 global loads
        v2f fa[4];
#pragma unroll
        for (int k = 0; k < 4; ++k)
            fa[k] = *(const v2f*)(f + lrow * 16 + 4 * k + 2 * lhalf);

        // GEMM1: T = F_c * Wx^T   (K=16 via 4 chained k=4 WMMAs)
        v8f t = {};
#pragma unroll
        for (int k = 0; k < 4; ++k)
            t = __builtin_amdgcn_wmma_f32_16x16x4_f32(false, fa[k], false, bWx[k],
                                                      (short)0, t, false, false);

        // D-layout -> B-layout: half-wave swap via shfl_xor(16) + selects
        const float t0 = t[0], t1 = t[1], t2 = t[2], t3 = t[3],
                    t4 = t[4], t5 = t[5], t6 = t[6], t7 = t[7];
        const float s0 = __shfl_xor(t0, 16, 32), s1 = __shfl_xor(t1, 16, 32);
        const float s2 = __shfl_xor(t2, 16, 32), s3 = __shfl_xor(t3, 16, 32);
        const float s4 = __shfl_xor(t4, 16, 32), s5 = __shfl_xor(t5, 16, 32);
        const float s6 = __shfl_xor(t6, 16, 32), s7 = __shfl_xor(t7, 16, 32);

        v2f tb[4];
        tb[0].x = lo ? t0 : s2;  tb[0].y = lo ? t1 : s3;   // rows 0,1 | 2,3
        tb[1].x = lo ? t4 : s6;  tb[1].y = lo ? t5 : s7;   // rows 4,5 | 6,7
        tb[2].x = lo ? s0 : t2;  tb[2].y = lo ? s1 : t3;   // rows 8,9 | 10,11
        tb[3].x = lo ? s4 : t6;  tb[3].y = lo ? s5 : t7;   // rows 12,13 | 14,15

        // GEMM2: Out = Wy * T
        v8f o = {};
#pragma unroll
        for (int k = 0; k < 4; ++k)
            o = __builtin_amdgcn_wmma_f32_16x16x4_f32(false, aWy[k], false, tb[k],
                                                      (short)0, o, false, false);

        // store: lane holds (py = j + 8*lhalf, px = lrow) -> coalesced 2x64B runs
        float* op = ob + (size_t)c * HW;
#pragma unroll
        for (int j = 0; j < 8; ++j) {
            const int m = j + lhalf * 8;
            op[m * 16 + lrow] = o[j];
        }
    }
}

// ---------------------------------------------------------------------------
extern "C" void kernel_launch(void* const* d_in, const int* in_sizes, int n_in,
                              void* d_out, int out_size, void* d_ws, size_t ws_size,
                              hipStream_t stream) {
    const float* feat = (const float*)d_in[0];   // (4,2048,8,16,16) f32
    const float* rois = (const float*)d_in[1];   // (32,5) f32
    float* out   = (float*)d_out;                          // (32,2048,1,16,16)
    float* featp = out + (size_t)R_ * C_ * HW;             // (4,2048,1,16,16)

    const int nvec = NB * C_ * HW / 4;   // 524288 float4 threads
    tmean_kernel<<<nvec / 256, 256, 0, stream>>>(feat, featp);
    roi_align_wmma_kernel<<<R_ * CB, 256, 0, stream>>>(featp, rois, out);
}